// SparsePooling_53437983096844
// MI455X (gfx1250) — compile-verified
//
#include <hip/hip_runtime.h>

// MoE top-3 routing + sparse expert GEMM for MI455X (gfx1250, wave32).
// out[b] = sum_k gate[b,k] * (x[b] @ We[e_k]),  gate = softmax over original
// logits of the 3 experts with highest (logit + bias).
// Sparse dispatch: 2*B*K*D*F = 51.5 GFLOP fp32 (5.3x less than reference's
// dense form). Working set fits L2 (192MB) -> compute/L2-bound; tiles sized
// to maximize WMMA fragment reuse from LDS.

#define B_TOK 16384
#define D_DIM 1024
#define E_NUM 16
#define F_DIM 512
#define TOPK  3

#define TILE_M 64
#define TILE_N 128
#define KC     32
#define NCHUNK (D_DIM / KC)
#define AS_STRIDE 36    // 16B-aligned rows, conflict-free column reads
#define BS_STRIDE 144   // 16B-aligned rows, conflict-free row-pair reads

typedef __attribute__((ext_vector_type(2))) float v2f;
typedef __attribute__((ext_vector_type(8))) float v8f;
typedef __attribute__((ext_vector_type(4))) int v4i;

// ---- gfx1250 async memory->LDS path (ASYNCcnt), guarded so compile is safe
#if defined(__has_builtin)
#if __has_builtin(__builtin_amdgcn_global_load_async_to_lds_b128)
#define USE_ASYNC_LDS 1
#endif
#endif
#ifndef USE_ASYNC_LDS
#define USE_ASYNC_LDS 0
#endif

#if USE_ASYNC_LDS
#if __has_builtin(__builtin_amdgcn_s_wait_asynccnt)
#define WAIT_ASYNC(n) __builtin_amdgcn_s_wait_asynccnt(n)
#else
#define WAIT_ASYNC(n) asm volatile("s_wait_asynccnt %0" ::"n"(n) : "memory")
#endif
typedef __attribute__((address_space(1))) v4i* gas_v4i_ptr;
typedef __attribute__((address_space(3))) v4i* las_v4i_ptr;
static __device__ __forceinline__ void cp_async16(const float* g, float* l) {
  __builtin_amdgcn_global_load_async_to_lds_b128(
      (gas_v4i_ptr)g, (las_v4i_ptr)l, /*imm_off=*/0, /*cpol=*/0);
}
#endif

// ---------------------------------------------------------------- zero init
__global__ __launch_bounds__(256) void moe_zero_kernel(float* __restrict__ out,
                                                       int n,
                                                       int* __restrict__ cnt) {
  int i = blockIdx.x * blockDim.x + threadIdx.x;
  if (i < n) out[i] = 0.0f;
  if (i < E_NUM) cnt[i] = 0;
}

// ---------------------------------------------------------------- gating
// One wave (32 lanes) per token: lanes split D, 16 logits in registers,
// wave32 shfl_xor reduction, lane 0 does top-3 + softmax + list append.
__global__ __launch_bounds__(256) void moe_gate_kernel(
    const float* __restrict__ x, const float* __restrict__ Wg,
    const float* __restrict__ bias_e, int* __restrict__ cnt,
    int* __restrict__ idx_list, float* __restrict__ wt_list) {
  const int lane = threadIdx.x & 31;
  const int b = blockIdx.x * (blockDim.x >> 5) + (threadIdx.x >> 5);
  if (b >= B_TOK) return;

  const float4* xr = (const float4*)(x + (size_t)b * D_DIM);

  float acc[E_NUM];
#pragma unroll
  for (int e = 0; e < E_NUM; ++e) acc[e] = 0.0f;

#pragma unroll
  for (int it = 0; it < 8; ++it) {
    const int q = it * 32 + lane;  // float4 index into x row (coalesced)
    const float4 xv = xr[q];
    const int d0 = q * 4;
    const float4* wr = (const float4*)(Wg + (size_t)d0 * E_NUM);
#pragma unroll
    for (int j = 0; j < 4; ++j) {
      const float xs = (j == 0) ? xv.x : (j == 1) ? xv.y : (j == 2) ? xv.z : xv.w;
#pragma unroll
      for (int e4 = 0; e4 < 4; ++e4) {
        const float4 w = wr[j * 4 + e4];
        acc[e4 * 4 + 0] += xs * w.x;
        acc[e4 * 4 + 1] += xs * w.y;
        acc[e4 * 4 + 2] += xs * w.z;
        acc[e4 * 4 + 3] += xs * w.w;
      }
    }
  }

#pragma unroll
  for (int e = 0; e < E_NUM; ++e) {
    float v = acc[e];
    v += __shfl_xor(v, 16, 32);
    v += __shfl_xor(v, 8, 32);
    v += __shfl_xor(v, 4, 32);
    v += __shfl_xor(v, 2, 32);
    v += __shfl_xor(v, 1, 32);
    acc[e] = v;
  }

  if (lane == 0) {
    float biased[E_NUM];
#pragma unroll
    for (int e = 0; e < E_NUM; ++e) biased[e] = acc[e] + bias_e[e];

    int eid[TOPK];
    float ov[TOPK];
    unsigned taken = 0u;
#pragma unroll
    for (int k = 0; k < TOPK; ++k) {
      float bv = -3.4e38f;
      int bi = 0;
      for (int e = 0; e < E_NUM; ++e) {
        if (!((taken >> e) & 1u) && biased[e] > bv) { bv = biased[e]; bi = e; }
      }
      taken |= 1u << bi;
      eid[k] = bi;
      ov[k] = acc[bi];  // softmax over ORIGINAL logits
    }
    const float mx = fmaxf(ov[0], fmaxf(ov[1], ov[2]));
    const float p0 = __expf(ov[0] - mx);
    const float p1 = __expf(ov[1] - mx);
    const float p2 = __expf(ov[2] - mx);
    const float inv = 1.0f / (p0 + p1 + p2);
    const float w[TOPK] = {p0 * inv, p1 * inv, p2 * inv};
#pragma unroll
    for (int k = 0; k < TOPK; ++k) {
      const int pos = atomicAdd(&cnt[eid[k]], 1);
      idx_list[eid[k] * B_TOK + pos] = b;
      wt_list[eid[k] * B_TOK + pos] = w[k];
    }
  }
}

// ---------------------------------------------------------------- expert GEMM
// grid = (256 m-tiles, 4 F-blocks, 16 experts); early-exit past cnt[e].
// Block: 256 threads = 8 waves as 2(m) x 4(n); each wave owns a 32x32 tile
// = 2x2 f32 16x16 WMMA accumulators (both frags reused twice). K-loop
// double-buffers A (gathered x rows) and B (We slice) through LDS, using
// GLOBAL_LOAD_ASYNC_TO_LDS_B128 + s_wait_asynccnt when available.
__global__ __launch_bounds__(256) void moe_gemm_kernel(
    const float* __restrict__ x, const float* __restrict__ We,
    const int* __restrict__ cnt, const int* __restrict__ idx_list,
    const float* __restrict__ wt_list, float* __restrict__ out) {
  const int e = blockIdx.z;
  const int fbase = blockIdx.y * TILE_N;
  const int mt = blockIdx.x;
  const int n_e = cnt[e];
  if (mt * TILE_M >= n_e) return;

  __shared__ float As[2][TILE_M][AS_STRIDE];
  __shared__ float Bs[2][KC][BS_STRIDE];
  __shared__ int tok_s[TILE_M];
  __shared__ float wt_s[TILE_M];

  const int tid = threadIdx.x;
  const int lane = tid & 31;
  const int halfl = lane >> 4;  // lane half selects K MSB pair
  const int l15 = lane & 15;
  const int wv = tid >> 5;
  const int wm = wv >> 2;  // 0..1 -> rows wm*32 .. +31
  const int wn = wv & 3;   // 0..3 -> cols wn*32 .. +31

  if (tid < TILE_M) {
    const int r = mt * TILE_M + tid;
    if (r < n_e) {
      tok_s[tid] = idx_list[e * B_TOK + r];
      wt_s[tid] = wt_list[e * B_TOK + r];
    } else {
      tok_s[tid] = 0;  // valid address, zero weight => contributes nothing
      wt_s[tid] = 0.0f;
    }
  }
  __syncthreads();

  // cooperative-load assignments
  const int a_row = tid >> 2;       // 64 rows, 4 threads/row, 8 floats each
  const int a_col = (tid & 3) * 8;
  const int b_row = tid >> 3;       // 32 rows, 8 threads/row, 16 floats each
  const int b_col = (tid & 7) * 16;

  const size_t xrow = (size_t)tok_s[a_row] * D_DIM;
  const float* wep = We + (size_t)e * D_DIM * F_DIM;

  v8f cacc[2][2] = {{{}, {}}, {{}, {}}};

  auto compute = [&](int buf) {
#pragma unroll
    for (int k4 = 0; k4 < KC / 4; ++k4) {
      const int kb = k4 * 4;
      // ISA 16x4 f32 A layout: VGPR idx = K LSB, lane half = K MSB.
      v2f af[2], bf[2];
#pragma unroll
      for (int ns = 0; ns < 2; ++ns) {
        bf[ns].x = Bs[buf][kb + 2 * halfl + 0][wn * 32 + ns * 16 + l15];
        bf[ns].y = Bs[buf][kb + 2 * halfl + 1][wn * 32 + ns * 16 + l15];
      }
#pragma unroll
      for (int ms = 0; ms < 2; ++ms) {
        af[ms].x = As[buf][wm * 32 + ms * 16 + l15][kb + 2 * halfl + 0];
        af[ms].y = As[buf][wm * 32 + ms * 16 + l15][kb + 2 * halfl + 1];
      }
#pragma unroll
      for (int ms = 0; ms < 2; ++ms)
#pragma unroll
        for (int ns = 0; ns < 2; ++ns)
          cacc[ms][ns] = __builtin_amdgcn_wmma_f32_16x16x4_f32(
              false, af[ms], false, bf[ns], (short)0, cacc[ms][ns], false,
              false);
    }
  };

#if USE_ASYNC_LDS
  auto issue = [&](int ci, int buf) {
    const int dc = ci * KC;
    const float* ga = x + xrow + dc + a_col;
    const float* gb = wep + (size_t)(dc + b_row) * F_DIM + fbase + b_col;
    cp_async16(ga, &As[buf][a_row][a_col]);
    cp_async16(ga + 4, &As[buf][a_row][a_col + 4]);
#pragma unroll
    for (int j = 0; j < 4; ++j)
      cp_async16(gb + 4 * j, &Bs[buf][b_row][b_col + 4 * j]);
  };

  issue(0, 0);
  for (int ci = 0; ci < NCHUNK; ++ci) {
    const int cur = ci & 1;
    if (ci + 1 < NCHUNK) {
      issue(ci + 1, cur ^ 1);  // prefetch next chunk into other buffer
      WAIT_ASYNC(6);           // the 6 in-order ops of chunk ci are done
    } else {
      WAIT_ASYNC(0);
    }
    __syncthreads();  // all waves' chunk-ci data visible in LDS
    compute(cur);
    __syncthreads();  // reads done before buffer is overwritten
  }
#else
  for (int ci = 0; ci < NCHUNK; ++ci) {
    const int dc = ci * KC;
    const float4 av0 = *(const float4*)(x + xrow + dc + a_col);
    const float4 av1 = *(const float4*)(x + xrow + dc + a_col + 4);
    float4 bv[4];
#pragma unroll
    for (int j = 0; j < 4; ++j)
      bv[j] = *(const float4*)(wep + (size_t)(dc + b_row) * F_DIM + fbase +
                               b_col + 4 * j);
    __syncthreads();
    *(float4*)&As[0][a_row][a_col] = av0;
    *(float4*)&As[0][a_row][a_col + 4] = av1;
#pragma unroll
    for (int j = 0; j < 4; ++j) *(float4*)&Bs[0][b_row][b_col + 4 * j] = bv[j];
    __syncthreads();
    compute(0);
  }
#endif

  // Scatter-add: C/D layout VGPR r: lanes0-15 -> M=r, lanes16-31 -> M=8+r.
#pragma unroll
  for (int ms = 0; ms < 2; ++ms) {
#pragma unroll
    for (int ns = 0; ns < 2; ++ns) {
      const int col = fbase + wn * 32 + ns * 16 + l15;
#pragma unroll
      for (int r = 0; r < 8; ++r) {
        const int m_local = wm * 32 + ms * 16 + r + halfl * 8;
        atomicAdd(out + (size_t)tok_s[m_local] * F_DIM + col,
                  wt_s[m_local] * cacc[ms][ns][r]);
      }
    }
  }
}

// ---------------------------------------------------------------- launcher
extern "C" void kernel_launch(void* const* d_in, const int* in_sizes, int n_in,
                              void* d_out, int out_size, void* d_ws,
                              size_t ws_size, hipStream_t stream) {
  const float* x = (const float*)d_in[0];
  const float* Wg = (const float*)d_in[1];
  const float* We = (const float*)d_in[2];
  const float* bias_e = (const float*)d_in[3];
  float* out = (float*)d_out;

  // workspace: [cnt: 16 ints | pad to 256B][idx_list: E*B ints][wt_list: E*B f32]
  int* cnt = (int*)d_ws;
  int* idx_list = (int*)((char*)d_ws + 256);
  float* wt_list =
      (float*)((char*)d_ws + 256 + sizeof(int) * (size_t)E_NUM * B_TOK);

  const int n_out = B_TOK * F_DIM;
  moe_zero_kernel<<<(n_out + 255) / 256, 256, 0, stream>>>(out, n_out, cnt);

  moe_gate_kernel<<<B_TOK / 8, 256, 0, stream>>>(x, Wg, bias_e, cnt, idx_list,
                                                 wt_list);

  dim3 grid(B_TOK / TILE_M, F_DIM / TILE_N, E_NUM);
  moe_gemm_kernel<<<grid, 256, 0, stream>>>(x, We, cnt, idx_list, wt_list, out);
}